// QNetwork_28561532518999
// MI455X (gfx1250) — compile-verified
//
#include <hip/hip_runtime.h>
#include <hip/hip_bf16.h>

#define NUM_NODES   5000
#define NUM_COLORS  20
#define NODE_FEAT   4
#define COLOR_FEAT  3
#define COMBINED    7
#define HIDDEN      128
#define N_GRAPH     (NUM_NODES * NUM_COLORS)   // 100000
#define LDS_STRIDE  132                        // 128 + 4 pad -> conflict-free b64 reads

typedef __attribute__((ext_vector_type(2))) float v2f;
typedef __attribute__((ext_vector_type(8))) float v8f;

// ---------------------------------------------------------------------------
// Small utility kernels
// ---------------------------------------------------------------------------
__global__ void fill_f32(float* __restrict__ p, float v, int n) {
    int i = blockIdx.x * blockDim.x + threadIdx.x;
    if (i < n) p[i] = v;
}

// deg[dst[e]] += 1  (self-loop contribution is baked into the fill value 1.0)
__global__ void degree_kernel(const int* __restrict__ dst, float* __restrict__ deg, int E) {
    int i = blockIdx.x * blockDim.x + threadIdx.x;
    if (i < E) atomicAdd(&deg[dst[i]], 1.0f);
}

__global__ void rsqrt_inplace(float* __restrict__ p, int n) {
    int i = blockIdx.x * blockDim.x + threadIdx.x;
    if (i < n) p[i] = __frsqrt_rn(p[i]);       // deg >= 1 always (self loops)
}

// combined[i, 0:4] = node_feat[i/20], combined[i, 4:7] = color_feat[i%20]
__global__ void build_combined(const float* __restrict__ flat, float* __restrict__ out) {
    int i = blockIdx.x * blockDim.x + threadIdx.x;
    if (i >= N_GRAPH) return;
    int node  = i / NUM_COLORS;
    int color = i - node * NUM_COLORS;
    const float* nf = flat + 3 + node * NODE_FEAT;
    const float* cf = flat + 3 + NUM_NODES * NODE_FEAT + color * COLOR_FEAT;
    float* o = out + (size_t)i * COMBINED;
    o[0] = nf[0]; o[1] = nf[1]; o[2] = nf[2]; o[3] = nf[3];
    o[4] = cf[0]; o[5] = cf[1]; o[6] = cf[2];
}

// XW[row, col] = sum_k combined[row,k] * W1[k,col]   (K = 7, VALU is right here)
__global__ void __launch_bounds__(128) gemm_k7(const float* __restrict__ C,
                                               const float* __restrict__ W1,
                                               float* __restrict__ XW) {
    int row = blockIdx.x;                // 0..N_GRAPH-1
    int col = threadIdx.x;               // 0..127
    const float* c = C + (size_t)row * COMBINED;
    float acc = 0.0f;
#pragma unroll
    for (int k = 0; k < COMBINED; ++k)
        acc = fmaf(c[k], W1[k * HIDDEN + col], acc);
    XW[(size_t)row * HIDDEN + col] = acc;
}

// H[dst] += norm * XW[src]  over explicit edges (L2-resident atomic scatter)
#define EDGES_PER_BLOCK 8
__global__ void __launch_bounds__(128) scatter_edges(const int* __restrict__ src,
                                                     const int* __restrict__ dst,
                                                     const float* __restrict__ dinv,
                                                     const float* __restrict__ XW,
                                                     float* __restrict__ H, int E) {
    int c  = threadIdx.x;                          // channel 0..127
    int e0 = blockIdx.x * EDGES_PER_BLOCK;
#pragma unroll
    for (int j = 0; j < EDGES_PER_BLOCK; ++j) {
        int e = e0 + j;
        if (e >= E) return;
        int s = src[e], d = dst[e];
        float nrm = dinv[s] * dinv[d];
        atomicAdd(&H[(size_t)d * HIDDEN + c], XW[(size_t)s * HIDDEN + c] * nrm);
    }
}

// H = relu(H + XW * dinv^2 (self loop) + bias)
__global__ void __launch_bounds__(128) finalize_conv(const float* __restrict__ XW,
                                                     const float* __restrict__ dinv,
                                                     const float* __restrict__ bias,
                                                     float* __restrict__ H) {
    int i = blockIdx.x;                  // node
    int c = threadIdx.x;                 // channel
    float di = dinv[i];
    size_t idx = (size_t)i * HIDDEN + c;
    float v = H[idx] + XW[idx] * di * di + bias[c];
    H[idx] = fmaxf(v, 0.0f);
}

// ---------------------------------------------------------------------------
// 100000x128 @ 128x128 GEMM using V_WMMA_F32_16X16X4_F32 (fp32 WMMA).
// Block = 256 threads = 8 waves, one 16-row M tile per block.
// The 16x128 X tile (8 KB, contiguous in memory) is staged into LDS once per
// block with CDNA5 async loads (global_load_async_to_lds_b128, ASYNCcnt).
// The LDS destination address is derived from the generic pointer (low 32
// bits of a generic LDS pointer == LDS byte offset, ISA 10.2), which also
// escapes `tile` so the compiler must honor the asm's memory clobber and
// keep the ds_load_b64 A-fragment reads. LDS rows padded to 132 floats ->
// conflict-free b64 reads. Wave w computes N tile w: 32 chained WMMAs, K=128.
// ---------------------------------------------------------------------------
__global__ void __launch_bounds__(256) wmma_gemm128(const float* __restrict__ X,
                                                    const float* __restrict__ W,
                                                    const float* __restrict__ bias,
                                                    float* __restrict__ Y,
                                                    int relu) {
    __shared__ float tile[16 * LDS_STRIDE];      // 8448 B

    const int lane  = threadIdx.x & 31;
    const int wv    = threadIdx.x >> 5;          // N tile 0..7
    const int row0  = blockIdx.x * 16;           // M tile base
    const int col0  = wv * 16;
    const int l15   = lane & 15;
    const int khalf = (lane >> 4) * 2;           // 0 or 2
    const int ncol  = col0 + l15;

    // ---- async cooperative copy: 512 float4s, 2 per thread, coalesced ----
    {
        const float* gbase = X + (size_t)row0 * HIDDEN;
        unsigned t = threadIdx.x;
#pragma unroll
        for (int chunk = 0; chunk < 2; ++chunk) {
            unsigned f4  = t + chunk * 256;                // float4 index 0..511
            unsigned row = f4 >> 5;                        // 32 float4s per row
            unsigned c4  = f4 & 31;
            // low 32 bits of generic LDS pointer == LDS byte offset (VDST value);
            // ptrtoint also escapes `tile` so its contents aren't treated as undef
            unsigned ldsb = (unsigned)(uintptr_t)(tile + row * LDS_STRIDE + c4 * 4);
            unsigned long long gaddr =
                (unsigned long long)(uintptr_t)(gbase + (size_t)f4 * 4);
            asm volatile("global_load_async_to_lds_b128 %0, %1, off"
                         :: "v"(ldsb), "v"(gaddr)
                         : "memory");
        }
        asm volatile("s_wait_asynccnt 0x0" ::: "memory");
        __syncthreads();
    }

    const float* __restrict__ arow = tile + l15 * LDS_STRIDE;

    v8f c = {0.f, 0.f, 0.f, 0.f, 0.f, 0.f, 0.f, 0.f};
#pragma unroll
    for (int k = 0; k < HIDDEN; k += 4) {
        v2f a, b;
        a.x = arow[k + khalf];                       // ds_load_b64, conflict-free
        a.y = arow[k + khalf + 1];
        b.x = W[(size_t)(k + khalf)     * HIDDEN + ncol];   // coalesced 64B rows
        b.y = W[(size_t)(k + khalf + 1) * HIDDEN + ncol];
        // 8 args: (neg_a, A, neg_b, B, c_mod, C, reuse_a, reuse_b)
        c = __builtin_amdgcn_wmma_f32_16x16x4_f32(false, a, false, b,
                                                  (short)0, c, false, false);
    }

    const float bv    = bias ? bias[ncol] : 0.0f;
    const int   rbase = row0 + ((lane >> 4) << 3);   // +8 for upper half-wave
#pragma unroll
    for (int r = 0; r < 8; ++r) {
        float v = c[r] + bv;
        if (relu) v = fmaxf(v, 0.0f);
        Y[(size_t)(rbase + r) * HIDDEN + ncol] = v;
    }
}

// q[row] = dot(H[row,:], w) + b   — one wave32 per row, shuffle reduction
__global__ void __launch_bounds__(256) fc2_kernel(const float* __restrict__ H,
                                                  const float* __restrict__ w,
                                                  const float* __restrict__ b,
                                                  float* __restrict__ out, int M) {
    int gid  = blockIdx.x * blockDim.x + threadIdx.x;
    int row  = gid >> 5;
    int lane = gid & 31;
    if (row >= M) return;
    const float* h = H + (size_t)row * HIDDEN;
    float s = 0.0f;
#pragma unroll
    for (int j = 0; j < 4; ++j)
        s = fmaf(h[lane + 32 * j], w[lane + 32 * j], s);
#pragma unroll
    for (int off = 16; off > 0; off >>= 1)
        s += __shfl_xor(s, off, 32);
    if (lane == 0) out[row] = s + b[0];
}

// ---------------------------------------------------------------------------
extern "C" void kernel_launch(void* const* d_in, const int* in_sizes, int n_in,
                              void* d_out, int out_size, void* d_ws, size_t ws_size,
                              hipStream_t stream) {
    const float* flat  = (const float*)d_in[0];
    const int*   ei    = (const int*)  d_in[1];   // [2, E]: row0 = src, row1 = dst
    const float* W1    = (const float*)d_in[2];
    const float* b1    = (const float*)d_in[3];
    const float* W2    = (const float*)d_in[4];
    const float* b2    = (const float*)d_in[5];
    const float* fc1w  = (const float*)d_in[6];
    const float* fc1b  = (const float*)d_in[7];
    const float* fc2w  = (const float*)d_in[8];
    const float* fc2b  = (const float*)d_in[9];
    float*       out   = (float*)d_out;

    const int E = in_sizes[1] / 2;                 // 1,600,000
    const int* src = ei;
    const int* dst = ei + E;

    // Workspace layout (floats)
    float* ws       = (float*)d_ws;
    float* combined = ws;                          //   700,000
    float* dinv     = ws + 700000;                 //   100,000 (deg, then rsqrt)
    float* bufA     = ws + 800000;                 // 12,800,000
    float* bufB     = bufA + (size_t)N_GRAPH * HIDDEN;
    float* bufC     = bufB + (size_t)N_GRAPH * HIDDEN;
    const int NH    = N_GRAPH * HIDDEN;            // 12,800,000

    const int T = 256;
    // 1) combined features
    build_combined<<<(N_GRAPH + T - 1) / T, T, 0, stream>>>(flat, combined);
    // 2) degree (init 1.0 for self loop) -> dinv = rsqrt(deg)
    fill_f32<<<(N_GRAPH + T - 1) / T, T, 0, stream>>>(dinv, 1.0f, N_GRAPH);
    degree_kernel<<<(E + T - 1) / T, T, 0, stream>>>(dst, dinv, E);
    rsqrt_inplace<<<(N_GRAPH + T - 1) / T, T, 0, stream>>>(dinv, N_GRAPH);
    // 3) layer 1: xw1 = combined @ W1 (K=7) -> aggregate -> relu(+b1)
    gemm_k7<<<N_GRAPH, 128, 0, stream>>>(combined, W1, bufA);
    fill_f32<<<(NH + T - 1) / T, T, 0, stream>>>(bufB, 0.0f, NH);
    scatter_edges<<<(E + EDGES_PER_BLOCK - 1) / EDGES_PER_BLOCK, 128, 0, stream>>>(
        src, dst, dinv, bufA, bufB, E);
    finalize_conv<<<N_GRAPH, 128, 0, stream>>>(bufA, dinv, b1, bufB);   // bufB = h1
    // 4) layer 2: xw2 = h1 @ W2 (WMMA) -> aggregate -> relu(+b2)
    wmma_gemm128<<<N_GRAPH / 16, 256, 0, stream>>>(bufB, W2, nullptr, bufA, 0);
    fill_f32<<<(NH + T - 1) / T, T, 0, stream>>>(bufC, 0.0f, NH);
    scatter_edges<<<(E + EDGES_PER_BLOCK - 1) / EDGES_PER_BLOCK, 128, 0, stream>>>(
        src, dst, dinv, bufA, bufC, E);
    finalize_conv<<<N_GRAPH, 128, 0, stream>>>(bufA, dinv, b2, bufC);   // bufC = h2
    // 5) fc1 (WMMA, fused bias+relu): h3 = relu(h2 @ fc1_w + fc1_b)
    wmma_gemm128<<<N_GRAPH / 16, 256, 0, stream>>>(bufC, fc1w, fc1b, bufB, 1);
    // 6) fc2: q = h3 @ fc2_w + fc2_b
    fc2_kernel<<<(N_GRAPH * 32 + T - 1) / T, T, 0, stream>>>(bufB, fc2w, fc2b, out, N_GRAPH);
}